// Filter_9723805958232
// MI455X (gfx1250) — compile-verified
//
#include <hip/hip_runtime.h>
#include <math.h>

// Problem geometry (reference: B=16384, N=4096, 10 cells)
#define NCOL   4096
#define NMASK  4095
#define HWIN   100        // half-window of the largest cell (c=100)
#define NF     10

typedef __attribute__((ext_vector_type(2))) float v2f;
typedef __attribute__((ext_vector_type(8))) float v8f;

// ---------------------------------------------------------------------------
// Kernel 1: per-row strict-local-max argmax (the 256 MB bandwidth pass).
// One block per row; row staged in LDS via coalesced float4 loads.
// ---------------------------------------------------------------------------
__global__ __launch_bounds__(256) void peak_argmax_kernel(
    const float* __restrict__ inp, int* __restrict__ top1)
{
    __shared__ float row[NCOL];
    __shared__ float bval[256];
    __shared__ int   bidx[256];

    const int r = blockIdx.x;
    const int t = threadIdx.x;

    const float4* src = (const float4*)(inp + (size_t)r * NCOL);
    float4*       dst = (float4*)row;
#pragma unroll
    for (int i = 0; i < 4; ++i)                    // 4 * 256 = 1024 float4 = 4096 floats
        dst[t + 256 * i] = src[t + 256 * i];
    __syncthreads();

    // each thread scans 16 contiguous elements; strict local max, keep first best
    float bv = -INFINITY;
    int   bi = 0;
    const int base = t * 16;
#pragma unroll
    for (int j = 0; j < 16; ++j) {
        int i = base + j;
        if (i >= 1 && i <= NCOL - 2) {
            float x = row[i];
            if (x > row[i - 1] && x > row[i + 1]) {
                if (x > bv) { bv = x; bi = i; }    // strict > keeps lowest index on tie
            }
        }
    }
    bval[t] = bv;
    bidx[t] = bi;
    __syncthreads();

    // tree reduce, tie-break toward lower index (jnp.argmax first-occurrence)
    for (int s = 128; s > 0; s >>= 1) {
        if (t < s) {
            float ov = bval[t + s];
            int   oi = bidx[t + s];
            if (ov > bval[t] || (ov == bval[t] && oi < bidx[t])) {
                bval[t] = ov; bidx[t] = oi;
            }
        }
        __syncthreads();
    }
    if (t == 0) top1[r] = bidx[0];                  // all -inf -> 0, matches argmax
}

// ---------------------------------------------------------------------------
// Kernel 2: per-row softmax-weighted DoA features, 10 cells = 10 waves.
// Largest window (200 elems, circular) staged once in LDS; each wave32
// computes one cell over its sub-range with shuffle reductions.
// ---------------------------------------------------------------------------
__global__ __launch_bounds__(320) void feature_kernel(
    const float* __restrict__ inp,
    const float* __restrict__ ss_g,
    const int*   __restrict__ top1,
    float*       __restrict__ feat)
{
    __shared__ float vals[2 * HWIN];
    __shared__ float ssv [2 * HWIN];

    const int r = blockIdx.x;
    const int t = threadIdx.x;
    const int peak = top1[r];

    if (t < 2 * HWIN) {
        int idx = (peak - HWIN + t) & NMASK;        // circular window
        vals[t] = inp[(size_t)r * NCOL + idx];
        ssv [t] = ss_g[idx];
    }
    __syncthreads();

    const int wv   = t >> 5;                        // wave id = cell id (0..9)
    const int lane = t & 31;
    const int cells[NF] = {2, 12, 23, 34, 45, 56, 67, 78, 89, 100};
    const int c    = cells[wv];
    const int base = HWIN - c;
    const int len  = 2 * c;

    // stable softmax: max
    float m = -INFINITY;
    for (int e = lane; e < len; e += 32) m = fmaxf(m, vals[base + e]);
#pragma unroll
    for (int off = 16; off > 0; off >>= 1) m = fmaxf(m, __shfl_xor(m, off, 32));

    // sum(exp) and sum(exp * search_space)
    float se = 0.f, ses = 0.f;
    for (int e = lane; e < len; e += 32) {
        float ex = expf(vals[base + e] - m);
        se  += ex;
        ses += ex * ssv[base + e];
    }
#pragma unroll
    for (int off = 16; off > 0; off >>= 1) {
        se  += __shfl_xor(se,  off, 32);
        ses += __shfl_xor(ses, off, 32);
    }
    if (lane == 0) feat[(size_t)r * NF + wv] = ses / se;
}

// ---------------------------------------------------------------------------
// Kernel 3: out = relu(feat[B,10] @ w[10,1] + b) via V_WMMA_F32_16X16X4_F32.
// Each wave handles 16 rows; K=10 done as 3 chained K=4 WMMAs (zero-padded).
// B-matrix replicated across all 16 N columns -> every column of D holds the
// dot product; lanes with N==0 (lane 0 / lane 16) write 8 rows each.
// A-frag layout (ISA 32-bit A 16x4): lanes 0-15 M=lane, V0=K0 V1=K1;
// lanes 16-31 M=lane-16, V0=K2 V1=K3.
// ---------------------------------------------------------------------------
__global__ __launch_bounds__(128) void gemv_wmma_kernel(
    const float* __restrict__ feat,
    const float* __restrict__ w,
    const float* __restrict__ bias,
    float*       __restrict__ out)
{
    const int t       = threadIdx.x;
    const int wv      = t >> 5;
    const int lane    = t & 31;
    const int rowBase = (blockIdx.x * 4 + wv) * 16;
    const int m       = lane & 15;                  // M within tile
    const int kb      = (lane >> 4) << 1;           // 0 (lanes 0-15) or 2 (16-31)

    v8f c = {};
#pragma unroll
    for (int kc = 0; kc < 3; ++kc) {
        const int k0 = kc * 4 + kb;                 // K of VGPR0; VGPR1 = k0+1
        v2f a, bm;
        a.x  = (k0     < NF) ? feat[(size_t)(rowBase + m) * NF + k0]     : 0.f;
        a.y  = (k0 + 1 < NF) ? feat[(size_t)(rowBase + m) * NF + k0 + 1] : 0.f;
        bm.x = (k0     < NF) ? w[k0]     : 0.f;     // replicate w across all N
        bm.y = (k0 + 1 < NF) ? w[k0 + 1] : 0.f;
        c = __builtin_amdgcn_wmma_f32_16x16x4_f32(
                /*neg_a=*/false, a, /*neg_b=*/false, bm,
                /*c_mod=*/(short)0, c, /*reuse_a=*/false, /*reuse_b=*/false);
    }

    const float bv = bias[0];
    if (m == 0) {                                   // N==0 lanes: 0 and 16
        const int mb = (lane >> 4) * 8;             // rows 0-7 or 8-15
#pragma unroll
        for (int rr = 0; rr < 8; ++rr) {
            float v = c[rr] + bv;
            out[rowBase + mb + rr] = v > 0.f ? v : 0.f;
        }
    }
}

// ---------------------------------------------------------------------------
extern "C" void kernel_launch(void* const* d_in, const int* in_sizes, int n_in,
                              void* d_out, int out_size, void* d_ws, size_t ws_size,
                              hipStream_t stream)
{
    const float* inp = (const float*)d_in[0];       // [B, 4096] f32
    const float* ss  = (const float*)d_in[1];       // [4096]    f32
    const float* w   = (const float*)d_in[2];       // [1, 10]   f32
    const float* b   = (const float*)d_in[3];       // [1]       f32

    const int N = in_sizes[1];                      // 4096
    const int B = in_sizes[0] / N;                  // 16384

    int*   top1 = (int*)d_ws;                                   // B ints
    float* feat = (float*)((char*)d_ws + (size_t)B * sizeof(int)); // B*10 floats
    float* out  = (float*)d_out;                                // [B,1]

    peak_argmax_kernel<<<B, 256, 0, stream>>>(inp, top1);
    feature_kernel    <<<B, 320, 0, stream>>>(inp, ss, top1, feat);
    gemv_wmma_kernel  <<<B / 64, 128, 0, stream>>>(feat, w, b, out);
}